// TreeGNN_15960098472361
// MI455X (gfx1250) — compile-verified
//
#include <hip/hip_runtime.h>
#include <hip/hip_bf16.h>

#define HID 128
#define INDIM 417
#define NCLS 3
#define TK 32
#define EPS 1e-5f

typedef float v2f __attribute__((ext_vector_type(2)));
typedef float v8f __attribute__((ext_vector_type(8)));

// ---------------------------------------------------------------------------
// Dense GEMM  C[M x 128] = A[M x K] @ W[K x 128]  using fp32 WMMA 16x16x4.
// 256 threads = 8 waves per block; block computes a 128x128 tile.
// LDS layouts are K-pair-packed so every WMMA fragment is ONE aligned
// ds_load_b64 (no VGPR repacking between DS loads and v_wmma).
// ---------------------------------------------------------------------------
__global__ __launch_bounds__(256) void gemm_wmma_f32(
    const float* __restrict__ A, const float* __restrict__ W,
    float* __restrict__ C, int M, int K)
{
    // As[r][k]: row-major, row stride 34 dwords (even -> v2f stays 8B-aligned,
    // 17*m mod 32 distinct for m=0..15 -> conflict-free column-pair reads)
    __shared__ float As[128][TK + 2];
    // Wp[kp][n][2] = { W[2kp][n], W[2kp+1][n] }  (K-pair packed, stride 258)
    __shared__ float Wp[TK / 2][2 * HID + 2];

    const int tid   = threadIdx.x;
    const int wave  = tid >> 5;         // 0..7
    const int lane  = tid & 31;
    const int m16   = lane & 15;
    const int g     = lane >> 4;        // half-wave select (ISA A/B/C layouts)
    const int row0  = blockIdx.x * 128;
    const int strip = wave * 16;

    v8f acc[8] = {};                    // 8 column tiles of 16

    for (int kk = 0; kk < K; kk += TK) {
        // Stage A chunk: rows row0..row0+127, cols kk..kk+31 (coalesced)
        #pragma unroll
        for (int i = tid; i < 128 * TK; i += 256) {
            int r = i >> 5, c = i & 31;
            int gr = row0 + r, gc = kk + c;
            As[r][c] = (gr < M && gc < K) ? A[(long long)gr * K + gc] : 0.0f;
        }
        // Stage W chunk K-pair packed: element (r, c) -> Wp[r>>1][2c + (r&1)]
        #pragma unroll
        for (int i = tid; i < TK * HID; i += 256) {
            int r = i >> 7, c = i & 127;
            int gr = kk + r;
            Wp[r >> 1][2 * c + (r & 1)] = (gr < K) ? W[gr * HID + c] : 0.0f;
        }
        __syncthreads();

        #pragma unroll
        for (int s = 0; s < TK / 4; ++s) {
            const int kb = s * 4 + g * 2;            // even
            const int kp = s * 2 + g;                // K-pair index
            // A frag: lane holds A[m16][kb..kb+1] -> one ds_load_b64
            v2f a = *reinterpret_cast<const v2f*>(&As[strip + m16][kb]);
            #pragma unroll
            for (int t = 0; t < 8; ++t) {
                // B frag: lane holds W[kb..kb+1][t*16+m16] -> one ds_load_b64
                v2f b = *reinterpret_cast<const v2f*>(&Wp[kp][2 * (t * 16 + m16)]);
                acc[t] = __builtin_amdgcn_wmma_f32_16x16x4_f32(
                    false, a, false, b, (short)0, acc[t], false, false);
            }
        }
        __syncthreads();
    }

    // C/D layout: VGPR r -> row strip + g*8 + r, col t*16 + m16
    #pragma unroll
    for (int t = 0; t < 8; ++t) {
        #pragma unroll
        for (int r = 0; r < 8; ++r) {
            int row = row0 + strip + g * 8 + r;
            if (row < M) C[(long long)row * HID + t * 16 + m16] = acc[t][r];
        }
    }
}

// ---------------------------------------------------------------------------
// Degree handling (self-loops folded in analytically)
// ---------------------------------------------------------------------------
__global__ void deg_init(float* deg, int N) {
    int i = blockIdx.x * 256 + threadIdx.x;
    if (i < N) deg[i] = 1.0f;            // self-loop contributes 1 to every node
}
__global__ void deg_edges(const int* __restrict__ dst, float* deg, int E) {
    int e = blockIdx.x * 256 + threadIdx.x;
    if (e < E) atomicAdd(&deg[dst[e]], 1.0f);
}
__global__ void deg_finalize(float* deg, int N) {
    int i = blockIdx.x * 256 + threadIdx.x;
    if (i < N) {
        float d = deg[i];
        deg[i] = (d > 0.0f) ? rsqrtf(d) : 0.0f;
    }
}

// ---------------------------------------------------------------------------
// Aggregation: out[i] = bias + h[i]*dis[i]^2  (self loop), then edge scatter
// ---------------------------------------------------------------------------
__global__ void init_agg(const float* __restrict__ h, const float* __restrict__ b,
                         const float* __restrict__ dis, float* __restrict__ out, int N) {
    long long i = (long long)blockIdx.x * 256 + threadIdx.x;
    if (i >= (long long)N * HID) return;
    int r = (int)(i >> 7), f = (int)(i & 127);
    float w = dis[r];
    out[i] = b[f] + h[i] * w * w;
}

// one wave32 per edge; each lane owns 4 features (float4) -> 128 features
__global__ __launch_bounds__(256) void edge_agg(
    const float* __restrict__ h, const int* __restrict__ src,
    const int* __restrict__ dst, const float* __restrict__ dis,
    float* __restrict__ out, int E)
{
    int e = blockIdx.x * 8 + (threadIdx.x >> 5);
    if (e >= E) return;
    int lane = threadIdx.x & 31;
    int s = src[e], d = dst[e];
    float w = dis[s] * dis[d];
    const float4 hv = *reinterpret_cast<const float4*>(h + (long long)s * HID + lane * 4);
    float* o = out + (long long)d * HID + lane * 4;
    atomicAdd(o + 0, hv.x * w);
    atomicAdd(o + 1, hv.y * w);
    atomicAdd(o + 2, hv.z * w);
    atomicAdd(o + 3, hv.w * w);
}

// ---------------------------------------------------------------------------
// BatchNorm (train-mode, biased var) + ReLU, and column sums for pooling
// ---------------------------------------------------------------------------
__global__ void zero_f(float* p, int n) {
    int i = blockIdx.x * 256 + threadIdx.x;
    if (i < n) p[i] = 0.0f;
}

__global__ __launch_bounds__(128) void col_stats(
    const float* __restrict__ x, float* __restrict__ sum, float* __restrict__ sumsq,
    int N, int rows_per_block, int do_sq)
{
    int f = threadIdx.x;                 // one feature column per thread
    long long r0 = (long long)blockIdx.x * rows_per_block;
    long long r1 = r0 + rows_per_block; if (r1 > N) r1 = N;
    float s = 0.0f, q = 0.0f;
    for (long long r = r0; r < r1; ++r) {
        float v = x[r * HID + f];
        s += v; q += v * v;
    }
    atomicAdd(&sum[f], s);
    if (do_sq) atomicAdd(&sumsq[f], q);
}

__global__ void bn_relu(float* __restrict__ x, const float* __restrict__ sum,
                        const float* __restrict__ sumsq, const float* __restrict__ gamma,
                        const float* __restrict__ beta, int N) {
    long long i = (long long)blockIdx.x * 256 + threadIdx.x;
    if (i >= (long long)N * HID) return;
    int f = (int)(i & 127);
    float invN = 1.0f / (float)N;
    float mean = sum[f] * invN;
    float var  = sumsq[f] * invN - mean * mean;
    float xn = (x[i] - mean) * rsqrtf(var + EPS);
    float y = xn * gamma[f] + beta[f];
    x[i] = (y > 0.0f) ? y : 0.0f;
}

// ---------------------------------------------------------------------------
// Final pooled FC: out[c] = (colsum/N) . fcW[:,c] + fcb[c]
// ---------------------------------------------------------------------------
__global__ __launch_bounds__(32) void fc_out(
    const float* __restrict__ colsum, const float* __restrict__ fcW,
    const float* __restrict__ fcb, float* __restrict__ out, int N)
{
    int c = threadIdx.x;
    if (c >= NCLS) return;
    float invN = 1.0f / (float)N;
    float acc = fcb[c];
    for (int f = 0; f < HID; ++f)
        acc += (colsum[f] * invN) * fcW[f * NCLS + c];
    out[c] = acc;
}

// ---------------------------------------------------------------------------
// Host-side orchestration
// ---------------------------------------------------------------------------
extern "C" void kernel_launch(void* const* d_in, const int* in_sizes, int n_in,
                              void* d_out, int out_size, void* d_ws, size_t ws_size,
                              hipStream_t stream) {
    const float* x    = (const float*)d_in[0];
    const int*   ei   = (const int*)  d_in[1];   // [2, E] flat: row 0 = src, row 1 = dst
    const float* W1   = (const float*)d_in[2];
    const float* b1   = (const float*)d_in[3];
    const float* W2   = (const float*)d_in[4];
    const float* b2   = (const float*)d_in[5];
    const float* W3   = (const float*)d_in[6];
    const float* b3   = (const float*)d_in[7];
    const float* g1   = (const float*)d_in[8];
    const float* be1  = (const float*)d_in[9];
    const float* g2   = (const float*)d_in[10];
    const float* be2  = (const float*)d_in[11];
    const float* g3   = (const float*)d_in[12];
    const float* be3  = (const float*)d_in[13];
    const float* fcW  = (const float*)d_in[14];
    const float* fcb  = (const float*)d_in[15];
    float* out = (float*)d_out;

    const int N = in_sizes[0] / INDIM;
    const int E = in_sizes[1] / 2;
    const int* src = ei;
    const int* dst = ei + E;

    // workspace layout (floats)
    float* bufA  = (float*)d_ws;                       // GEMM output [N*HID]
    float* bufB  = bufA + (long long)N * HID;          // agg/BN buffer [N*HID]
    float* dis   = bufB + (long long)N * HID;          // deg^{-1/2} [N]
    float* stats = dis + N;                            // sum[128], sumsq[128]
    float* ssq   = stats + HID;

    const int NE   = (int)(((long long)N * HID + 255) / 256);
    const int NB   = (N + 255) / 256;
    const int EB   = (E + 255) / 256;
    const int EAGG = (E + 7) / 8;
    const int GMM  = (N + 127) / 128;
    const int SBLK = 512;
    const int RPB  = (N + SBLK - 1) / SBLK;

    // degree -> deg^{-1/2}
    deg_init<<<NB, 256, 0, stream>>>(dis, N);
    deg_edges<<<EB, 256, 0, stream>>>(dst, dis, E);
    deg_finalize<<<NB, 256, 0, stream>>>(dis, N);

    const float* layer_in = x;
    int K = INDIM;
    const float* Ws[3]  = { W1, W2, W3 };
    const float* bs[3]  = { b1, b2, b3 };
    const float* gs[3]  = { g1, g2, g3 };
    const float* bes[3] = { be1, be2, be3 };

    for (int l = 0; l < 3; ++l) {
        gemm_wmma_f32<<<GMM, 256, 0, stream>>>(layer_in, Ws[l], bufA, N, K);
        init_agg<<<NE, 256, 0, stream>>>(bufA, bs[l], dis, bufB, N);
        edge_agg<<<EAGG, 256, 0, stream>>>(bufA, src, dst, dis, bufB, E);
        zero_f<<<1, 256, 0, stream>>>(stats, 2 * HID);
        col_stats<<<SBLK, 128, 0, stream>>>(bufB, stats, ssq, N, RPB, 1);
        bn_relu<<<NE, 256, 0, stream>>>(bufB, stats, ssq, gs[l], bes[l], N);
        layer_in = bufB;
        K = HID;
    }

    // mean-pool + FC
    zero_f<<<1, 256, 0, stream>>>(stats, HID);
    col_stats<<<SBLK, 128, 0, stream>>>(bufB, stats, ssq, N, RPB, 0);
    fc_out<<<1, 32, 0, stream>>>(stats, fcW, fcb, out, N);
}